// ACTFunction_23811298689464
// MI455X (gfx1250) — compile-verified
//
#include <hip/hip_runtime.h>
#include <math.h>

#define BATCH     64
#define DK        256
#define FEATURES  512
#define NUM_TAPE  2048
#define TOPK      512
#define TPS       16
#define THRESH    4.0f

// d_out flat offsets (floats), reference return order:
// query(64x256), halting_prob(64), remainders(64), n_updates(64),
// score_mask(64x2048), token_sel(64x1x512)
#define OFF_Q     0
#define OFF_HP    16384
#define OFF_REM   16448
#define OFF_NUP   16512
#define OFF_MASK  16576
#define OFF_TOK   147648

typedef __attribute__((ext_vector_type(2))) float v2f;
typedef __attribute__((ext_vector_type(8))) float v8f;

// ---------------------------------------------------------------------------
// Kernel 1: scores[b,n] = sum_{d<256} query[b,d] * tape[b,n,d]
// One wave computes 16 tape rows via V_WMMA_F32_16X16X4_F32:
//   A(16x4) = tile of keys (A layout: VGPR0 lanes0-15=K0 / lanes16-31=K2,
//             VGPR1=K1/K3 -> one b64 load per lane at row, d0+2*half)
//   B(4x16) = query chunk broadcast across N -> every D column equals the
//             dot-product partial; accumulate 64 WMMAs over d.
// ---------------------------------------------------------------------------
__global__ __launch_bounds__(256) void scores_kernel(
    const float* __restrict__ query, const float* __restrict__ tape,
    float* __restrict__ scores) {
  __shared__ float qs[DK];
  const int b   = blockIdx.y;
  const int tid = threadIdx.x;
  if (tid < DK) qs[tid] = query[b * DK + tid];
  __syncthreads();

  const int wave = tid >> 5;
  const int lane = tid & 31;
  const int half = lane >> 4;      // 0: K pair (0,1), 1: K pair (2,3)
  const int m    = lane & 15;      // row within 16-row tile
  const int n0   = blockIdx.x * 128 + wave * 16;
  const float* __restrict__ Krow =
      tape + ((size_t)b * NUM_TAPE + (size_t)(n0 + m)) * FEATURES;

#if __has_builtin(__builtin_amdgcn_wmma_f32_16x16x4_f32)
  v8f acc = {0.f, 0.f, 0.f, 0.f, 0.f, 0.f, 0.f, 0.f};
#pragma unroll 4
  for (int d0 = 0; d0 < DK; d0 += 4) {
    v2f a  = *(const v2f*)(Krow + d0 + 2 * half);   // A operand (global b64)
    v2f qb = *(const v2f*)(qs + d0 + 2 * half);     // B operand (LDS b64, bcast)
    acc = __builtin_amdgcn_wmma_f32_16x16x4_f32(
        false, a, false, qb, (short)0, acc, false, false);
  }
  // D layout: VGPR r, lanes0-15 -> M=r ; lanes16-31 -> M=r+8 (all N equal)
  if (m == 0) {
    float* srow = scores + (size_t)b * NUM_TAPE + n0 + 8 * half;
#pragma unroll
    for (int r = 0; r < 8; ++r) srow[r] = acc[r];
  }
#else
  // VALU fallback: each lane does half a row's reduction, pairwise shfl-xor.
  float p = 0.f;
  const int dbase = half * 128;
#pragma unroll 4
  for (int d = 0; d < 128; d += 4) {
    v2f a0 = *(const v2f*)(Krow + dbase + d);
    v2f a1 = *(const v2f*)(Krow + dbase + d + 2);
    p += a0.x * qs[dbase + d] + a0.y * qs[dbase + d + 1] +
         a1.x * qs[dbase + d + 2] + a1.y * qs[dbase + d + 3];
  }
  p += __shfl_xor(p, 16, 32);
  if (half == 0) scores[(size_t)b * NUM_TAPE + n0 + m] = p;
#endif
}

// ---------------------------------------------------------------------------
// Kernel 2: per-batch top-512 (sorted desc, stable) via bitonic sort of
// packed u64 keys; softmax weights (NOTE: jnp.take flattening bug -> raw
// logits come from scores ROW 0 at each batch's own indices); entropy,
// halting scalars, score_mask update.
// ---------------------------------------------------------------------------
__global__ __launch_bounds__(512) void topk_kernel(
    const float* __restrict__ scores, const float* __restrict__ hp_in,
    const float* __restrict__ rem_in, const float* __restrict__ nup_in,
    const float* __restrict__ mask_in, float* __restrict__ out,
    int* __restrict__ idx_ws, float* __restrict__ w_ws) {
  __shared__ unsigned long long keys[NUM_TAPE];  // 16 KB
  __shared__ float red[512];
  const int b = blockIdx.x, tid = threadIdx.x;

  // Build sort keys: ascending in (-score), tie-break ascending index (stable)
  for (int n = tid; n < NUM_TAPE; n += 512) {
    float s = scores[(size_t)b * NUM_TAPE + n] -
              mask_in[(size_t)b * NUM_TAPE + n] * 1e9f;
    unsigned u = __float_as_uint(s);
    u = (u & 0x80000000u) ? ~u : (u | 0x80000000u);   // orderable ascending
    keys[n] = ((unsigned long long)(~u) << 32) | (unsigned)n;
  }
  __syncthreads();

  // Bitonic sort, ascending, 2048 elements, 512 threads (2 pairs each)
  for (int k = 2; k <= NUM_TAPE; k <<= 1) {
    for (int j = k >> 1; j > 0; j >>= 1) {
      for (int t = tid; t < NUM_TAPE / 2; t += 512) {
        int i   = ((t & ~(j - 1)) << 1) | (t & (j - 1));
        int ixj = i | j;
        bool asc = (i & k) == 0;
        unsigned long long a = keys[i], c = keys[ixj];
        if ((a > c) == asc) { keys[i] = c; keys[ixj] = a; }
      }
      __syncthreads();
    }
  }

  // Exactly one selected element per thread (TOPK == blockDim == 512)
  const int idx = (int)(unsigned)(keys[tid] & 0xFFFFFFFFull);
  const float raw = scores[idx];          // flattened-take quirk: row 0!
  const float logit = raw * 0.0625f;      // 1/sqrt(256)

  red[tid] = logit; __syncthreads();
  for (int s = 256; s > 0; s >>= 1) {
    if (tid < s) red[tid] = fmaxf(red[tid], red[tid + s]);
    __syncthreads();
  }
  const float mx = red[0]; __syncthreads();

  const float e = expf(logit - mx);
  red[tid] = e; __syncthreads();
  for (int s = 256; s > 0; s >>= 1) {
    if (tid < s) red[tid] += red[tid + s];
    __syncthreads();
  }
  const float sum = red[0]; __syncthreads();
  const float w = e / sum;

  idx_ws[b * TOPK + tid] = idx;
  w_ws[b * TOPK + tid]   = w;

  red[tid] = w * w; __syncthreads();
  for (int s = 256; s > 0; s >>= 1) {
    if (tid < s) red[tid] += red[tid + s];
    __syncthreads();
  }
  const float entropy = 1.0f - red[0]; __syncthreads();

  red[tid] = (tid < TPS) ? w : 0.0f; __syncthreads();
  for (int s = 256; s > 0; s >>= 1) {
    if (tid < s) red[tid] += red[tid + s];
    __syncthreads();
  }
  const float sumw = red[0]; __syncthreads();

  // score_mask: copy then +1 at selected (distinct) indices
  for (int n = tid; n < NUM_TAPE; n += 512)
    out[OFF_MASK + (size_t)b * NUM_TAPE + n] =
        mask_in[(size_t)b * NUM_TAPE + n];
  __syncthreads();
  out[OFF_MASK + (size_t)b * NUM_TAPE + idx] += 1.0f;

  if (tid == 0) {
    const float hp = hp_in[b];
    float still = (hp < THRESH) ? 1.0f : 0.0f;
    float nh = ((hp + sumw) >= THRESH ? 1.0f : 0.0f) * still;
    float st2 = still - nh;
    float hp1 = hp + sumw * st2;
    hp1 = hp1 + nh * (THRESH - hp1);
    out[OFF_HP  + b] = hp1;
    out[OFF_REM + b] = rem_in[b] + (nh + st2) * entropy;
    out[OFF_NUP + b] = nup_in[b] + st2 + nh;
  }
}

// ---------------------------------------------------------------------------
// Kernel 3: partial weighted gather-sums over 64-k chunks (64 batches x 8).
// ---------------------------------------------------------------------------
__global__ __launch_bounds__(256) void gather_kernel(
    const float* __restrict__ tape, const int* __restrict__ idx_ws,
    const float* __restrict__ w_ws, float* __restrict__ partials) {
  __shared__ int   sidx[64];
  __shared__ float sw[64];
  const int b = blockIdx.x, chunk = blockIdx.y, tid = threadIdx.x;
  if (tid < 64) {
    sidx[tid] = idx_ws[b * TOPK + chunk * 64 + tid];
    sw[tid]   = w_ws[b * TOPK + chunk * 64 + tid];
  }
  __syncthreads();
  float a0 = 0.f, a1 = 0.f;
  for (int kk = 0; kk < 64; ++kk) {
    const float* __restrict__ T =
        tape + ((size_t)b * NUM_TAPE + (size_t)sidx[kk]) * FEATURES;
    const float wv = sw[kk];
    a0 += wv * T[tid];
    a1 += wv * T[tid + 256];
  }
  float* P = partials + ((size_t)b * 8 + chunk) * FEATURES;
  P[tid] = a0;
  P[tid + 256] = a1;
}

// ---------------------------------------------------------------------------
// Kernel 4: reduce 8 partials -> token_sel; query_out = (query+keys)/2.
// ---------------------------------------------------------------------------
__global__ __launch_bounds__(512) void finalize_kernel(
    const float* __restrict__ query, const float* __restrict__ partials,
    float* __restrict__ out) {
  const int b = blockIdx.x, f = threadIdx.x;
  float s = 0.f;
#pragma unroll
  for (int c = 0; c < 8; ++c)
    s += partials[((size_t)b * 8 + c) * FEATURES + f];
  out[OFF_TOK + (size_t)b * FEATURES + f] = s;
  if (f < DK) out[OFF_Q + b * DK + f] = (query[b * DK + f] + s) * 0.5f;
}

// ---------------------------------------------------------------------------
extern "C" void kernel_launch(void* const* d_in, const int* in_sizes, int n_in,
                              void* d_out, int out_size, void* d_ws,
                              size_t ws_size, hipStream_t stream) {
  const float* query = (const float*)d_in[0];
  const float* hp    = (const float*)d_in[1];
  const float* rem   = (const float*)d_in[2];
  const float* nup   = (const float*)d_in[3];
  const float* mask  = (const float*)d_in[4];
  const float* tape  = (const float*)d_in[5];
  float* out = (float*)d_out;

  char* ws = (char*)d_ws;
  float* scores   = (float*)(ws);                    // 64*2048*4   = 512 KB
  int*   idx_ws   = (int*)  (ws + 524288);           // 64*512*4    = 128 KB
  float* w_ws     = (float*)(ws + 655360);           // 64*512*4    = 128 KB
  float* partials = (float*)(ws + 786432);           // 64*8*512*4  = 1 MB

  scores_kernel  <<<dim3(16, BATCH), 256, 0, stream>>>(query, tape, scores);
  topk_kernel    <<<BATCH, 512, 0, stream>>>(scores, hp, rem, nup, mask, out,
                                             idx_ws, w_ws);
  gather_kernel  <<<dim3(BATCH, 8), 256, 0, stream>>>(tape, idx_ws, w_ws,
                                                      partials);
  finalize_kernel<<<BATCH, 512, 0, stream>>>(query, partials, out);
}